// Net_7825430413944
// MI455X (gfx1250) — compile-verified
//
#include <hip/hip_runtime.h>

#define N_NODES   100000
#define N_EDGES   1600000
#define D_IN      128
#define HIDDEN    128
#define N_CLASSES 16
#define FEAT_DIM  384

typedef __attribute__((ext_vector_type(2))) float v2f;
typedef __attribute__((ext_vector_type(8))) float v8f;

// ---------------------------------------------------------------------------
// Zero the aggregation scratch buffer (float4 stores).
// ---------------------------------------------------------------------------
__global__ void zero_f4(float4* __restrict__ p, int n4) {
  int i = blockIdx.x * blockDim.x + threadIdx.x;
  if (i < n4) {
    float4 z; z.x = 0.f; z.y = 0.f; z.z = 0.f; z.w = 0.f;
    p[i] = z;
  }
}

// ---------------------------------------------------------------------------
// Edge scatter: agg[dst] += w * x[src].  One wave (32 lanes) per edge,
// each lane handles 4 of the 128 features (float4 load + 4 f32 atomics).
// src/dst/w indices are wave-uniform -> scalar loads.
// ---------------------------------------------------------------------------
__global__ void scatter_edges(const float* __restrict__ x, int ldx,
                              const int* __restrict__ src,
                              const int* __restrict__ dst,
                              const float* __restrict__ w,
                              float* __restrict__ agg, int n_edges) {
  int e = blockIdx.x * (blockDim.x >> 5) + (threadIdx.x >> 5);
  e = __builtin_amdgcn_readfirstlane(e);
  if (e >= n_edges) return;
  int lane = threadIdx.x & 31;
  int s = src[e];
  int d = dst[e];
  float wt = w[e];
  const float4* xv = (const float4*)(x + (size_t)s * ldx);
  float4 v = xv[lane];
  float* out = agg + (size_t)d * HIDDEN + lane * 4;
  atomicAdd(out + 0, v.x * wt);
  atomicAdd(out + 1, v.y * wt);
  atomicAdd(out + 2, v.z * wt);
  atomicAdd(out + 3, v.w * wt);
}

// ---------------------------------------------------------------------------
// Fused GraphConv GEMMs via f32 WMMA:
//   xout = relu( agg @ Wrel^T + xin @ Wroot^T + bias )
//
// One wave computes a full 16-row x 128-col output strip:
//   - A-panel (16x128) for the active input matrix is loaded ONCE into
//     registers (64 VGPRs/lane), done in two passes (rel, then root).
//   - All 8 N-tile accumulators (8 x v8f = 64 VGPRs) stay live across
//     both passes, so activations are never re-read from memory.
//   - Weights (cache-resident, 64 KB each) stream from WGP$/L2.
//
// f32 16x16x4 A layout: lanes 0-15 hold K=k0..k0+1 of row M=lane,
// lanes 16-31 hold K=k0+2..k0+3 of row M=lane-16  (contiguous float2).
// B[k][n] = W[n][k] with W row-major -> same float2 pattern on W rows.
// C/D: VGPR r = rows {r, r+8} x 16 columns across lane halves.
// ---------------------------------------------------------------------------
__global__ void graphconv_wmma(const float* __restrict__ xin, int lda,
                               const float* __restrict__ agg,
                               const float* __restrict__ Wrel,
                               const float* __restrict__ Wroot,
                               const float* __restrict__ bias,
                               float* __restrict__ xout, int ldc, int n_nodes) {
  int wid = blockIdx.x * (blockDim.x >> 5) + (threadIdx.x >> 5);
  wid = __builtin_amdgcn_readfirstlane(wid);   // uniform guard, no EXEC games
  if (wid * 16 >= n_nodes) return;
  int lane = threadIdx.x & 31;
  int half = lane >> 4;                        // 0: K pair 0-1, 1: K pair 2-3
  int l    = lane & 15;
  int rowA = wid * 16 + l;

  v8f acc[8] = {};

  for (int pass = 0; pass < 2; ++pass) {
    const float* Abase = pass ? (xin + (size_t)rowA * lda)
                              : (agg + (size_t)rowA * HIDDEN);
    const float* Wbase = pass ? Wroot : Wrel;

    // Load this lane's slice of the 16x128 A-panel (32 float2 = 64 VGPRs).
    v2f a[32];
    #pragma unroll
    for (int ks = 0; ks < 32; ++ks)
      a[ks] = *(const v2f*)(Abase + ks * 4 + half * 2);

    #pragma unroll
    for (int t = 0; t < 8; ++t) {
      const float* Bp = Wbase + (size_t)(t * 16 + l) * HIDDEN + half * 2;
      #pragma unroll
      for (int ks = 0; ks < 32; ++ks) {
        v2f b = *(const v2f*)(Bp + ks * 4);
        acc[t] = __builtin_amdgcn_wmma_f32_16x16x4_f32(false, a[ks], false, b,
                                                       (short)0, acc[t],
                                                       false, false);
      }
    }
  }

  // bias + relu + store (row m, cols t*16 + l)
  #pragma unroll
  for (int t = 0; t < 8; ++t) {
    int n = t * 16 + l;
    float bv = bias[n];
    #pragma unroll
    for (int r = 0; r < 8; ++r) {
      int m = wid * 16 + half * 8 + r;
      float v = acc[t][r] + bv;
      xout[(size_t)m * ldc + n] = v > 0.f ? v : 0.f;
    }
  }
}

// ---------------------------------------------------------------------------
// Classifier: logits = log_softmax( feat @ Wlin^T + b_lin ).
// 16 classes == one 16-wide WMMA tile; K = 384 (96 WMMA steps).
// log-softmax is a 16-lane xor-shuffle reduction within each wave half
// (matches C layout: row m spans 16 lanes of one half).
// ---------------------------------------------------------------------------
__global__ void classifier_wmma(const float* __restrict__ feat,
                                const float* __restrict__ Wlin,
                                const float* __restrict__ blin,
                                float* __restrict__ logits, int n_nodes) {
  int wid = blockIdx.x * (blockDim.x >> 5) + (threadIdx.x >> 5);
  wid = __builtin_amdgcn_readfirstlane(wid);
  if (wid * 16 >= n_nodes) return;
  int lane = threadIdx.x & 31;
  int half = lane >> 4;
  int l    = lane & 15;

  const float* ap = feat + (size_t)(wid * 16 + l) * FEAT_DIM + half * 2;
  const float* bp = Wlin + (size_t)l * FEAT_DIM + half * 2;

  v8f c = {};
  #pragma unroll 8
  for (int k = 0; k < FEAT_DIM; k += 4) {
    v2f av = *(const v2f*)(ap + k);
    v2f bv = *(const v2f*)(bp + k);
    c = __builtin_amdgcn_wmma_f32_16x16x4_f32(false, av, false, bv,
                                              (short)0, c, false, false);
  }

  float bb = blin[l];
  #pragma unroll
  for (int r = 0; r < 8; ++r) {
    float v = c[r] + bb;
    float m = v;
    #pragma unroll
    for (int off = 1; off < 16; off <<= 1)
      m = fmaxf(m, __shfl_xor(m, off, 32));
    float e = __expf(v - m);
    float s = e;
    #pragma unroll
    for (int off = 1; off < 16; off <<= 1)
      s += __shfl_xor(s, off, 32);
    float out = v - m - __logf(s);
    int row = wid * 16 + half * 8 + r;
    logits[(size_t)row * N_CLASSES + l] = out;
  }
}

// ---------------------------------------------------------------------------
// Orchestration.  d_out = [logits 100000*16][features 100000*384].
// Layer outputs are written straight into their 128-col slice of features
// (stride 384) so the concat is free; next layer reads from there.
// d_ws holds the 100000x128 f32 aggregation scratch (51.2 MB).
// ---------------------------------------------------------------------------
extern "C" void kernel_launch(void* const* d_in, const int* in_sizes, int n_in,
                              void* d_out, int out_size, void* d_ws, size_t ws_size,
                              hipStream_t stream) {
  const float* x0     = (const float*)d_in[0];
  const int*   eidx   = (const int*)  d_in[1];
  const int*   src    = eidx;                 // edge_index[0]
  const int*   dst    = eidx + N_EDGES;       // edge_index[1]
  const float* w      = (const float*)d_in[2];
  const float* Wrel1  = (const float*)d_in[3];
  const float* b1     = (const float*)d_in[4];
  const float* Wroot1 = (const float*)d_in[5];
  const float* Wrel2  = (const float*)d_in[6];
  const float* b2     = (const float*)d_in[7];
  const float* Wroot2 = (const float*)d_in[8];
  const float* Wrel3  = (const float*)d_in[9];
  const float* b3     = (const float*)d_in[10];
  const float* Wroot3 = (const float*)d_in[11];
  const float* Wlin   = (const float*)d_in[12];
  const float* blin   = (const float*)d_in[13];

  float* logits = (float*)d_out;
  float* feat   = logits + (size_t)N_NODES * N_CLASSES;
  float* agg    = (float*)d_ws;

  const dim3 blk(256);                               // 8 waves per block
  const int  n4 = N_NODES * HIDDEN / 4;
  const dim3 zgrid((n4 + 255) / 256);
  const dim3 sgrid((N_EDGES + 7) / 8);               // 1 wave per edge
  const int  strips = (N_NODES + 15) / 16;           // 6250 16-row strips
  const dim3 cgrid((strips + 7) / 8);
  const dim3 fgrid((strips + 7) / 8);

  // Layer 1
  zero_f4<<<zgrid, blk, 0, stream>>>((float4*)agg, n4);
  scatter_edges<<<sgrid, blk, 0, stream>>>(x0, D_IN, src, dst, w, agg, N_EDGES);
  graphconv_wmma<<<cgrid, blk, 0, stream>>>(x0, D_IN, agg, Wrel1, Wroot1, b1,
                                            feat + 0, FEAT_DIM, N_NODES);
  // Layer 2
  zero_f4<<<zgrid, blk, 0, stream>>>((float4*)agg, n4);
  scatter_edges<<<sgrid, blk, 0, stream>>>(feat + 0, FEAT_DIM, src, dst, w, agg, N_EDGES);
  graphconv_wmma<<<cgrid, blk, 0, stream>>>(feat + 0, FEAT_DIM, agg, Wrel2, Wroot2, b2,
                                            feat + HIDDEN, FEAT_DIM, N_NODES);
  // Layer 3
  zero_f4<<<zgrid, blk, 0, stream>>>((float4*)agg, n4);
  scatter_edges<<<sgrid, blk, 0, stream>>>(feat + HIDDEN, FEAT_DIM, src, dst, w, agg, N_EDGES);
  graphconv_wmma<<<cgrid, blk, 0, stream>>>(feat + HIDDEN, FEAT_DIM, agg, Wrel3, Wroot3, b3,
                                            feat + 2 * HIDDEN, FEAT_DIM, N_NODES);
  // Classifier + log-softmax
  classifier_wmma<<<fgrid, blk, 0, stream>>>(feat, Wlin, blin, logits, N_NODES);
}